// Attn_66700842106908
// MI455X (gfx1250) — compile-verified
//
#include <hip/hip_runtime.h>
#include <math.h>

// ---------------------------------------------------------------------------
// Problem: attn = softmax_s( (enc @ W^T + b) . hidden ),  H=1024, S=32768.
// energy_s = enc_s . v + const, v = W^T hidden; const cancels in softmax.
// HBM-bound: 128 MB enc read -> ~5.5us floor @ 23.3 TB/s. FP32 WMMA GEMV.
// ---------------------------------------------------------------------------

#define H 1024
#define S 32768

typedef __attribute__((ext_vector_type(2))) float v2f;
typedef __attribute__((ext_vector_type(8))) float v8f;

// ---- Stage 1a: partial v = hidden @ W, split over 16 chunks of i ----------
// grid (H/256, 16), block 256.  part[chunk*H + h] = sum_{i in chunk} hid[i]*W[i,h]
__global__ __launch_bounds__(256) void k_vpart(const float* __restrict__ hidden,
                                               const float* __restrict__ W,
                                               float* __restrict__ part) {
    const int h  = blockIdx.x * 256 + threadIdx.x;
    const int i0 = blockIdx.y * (H / 16);
    float acc = 0.0f;
#pragma unroll 8
    for (int i = 0; i < H / 16; ++i) {
        acc = fmaf(hidden[i0 + i], W[(size_t)(i0 + i) * H + h], acc);
    }
    part[blockIdx.y * H + h] = acc;
}

// ---- Stage 1b: reduce the 16 partials -> v[h] -----------------------------
// grid H/256, block 256.
__global__ __launch_bounds__(256) void k_vreduce(const float* __restrict__ part,
                                                 float* __restrict__ v) {
    const int h = blockIdx.x * 256 + threadIdx.x;
    float acc = 0.0f;
#pragma unroll
    for (int j = 0; j < 16; ++j) acc += part[j * H + h];
    v[h] = acc;
}

// ---- Stage 2: energies = enc @ v via V_WMMA_F32_16X16X4_F32 ---------------
// grid S/(8*16) = 256 blocks, block 256 (8 waves). Each wave owns 16 seq rows.
// A (16x4 f32): lanes 0-15 -> M=lane, K={0,1}; lanes 16-31 -> M=lane-16, K={2,3}.
// B (4x16): every column identical (= v chunk), loaded from LDS; per-lane K
// assignment mirrors A.  C accumulates over 256 k-steps of 4.
__global__ __launch_bounds__(256) void k_energy(const float* __restrict__ enc,
                                                const float* __restrict__ v,
                                                float* __restrict__ energies) {
    __shared__ float vs[H];
    for (int i = threadIdx.x; i < H; i += 256) vs[i] = v[i];
    __syncthreads();

    const int wave = threadIdx.x >> 5;
    const int lane = threadIdx.x & 31;
    const int s0   = (blockIdx.x * 8 + wave) * 16;
    const int row  = lane & 15;
    const int hi   = lane >> 4;  // 0: K=0,1  1: K=2,3

    const float* arow = enc + (size_t)(s0 + row) * H + 2 * hi;
    const float* brow = vs + 2 * hi;

    v8f c = {0.f, 0.f, 0.f, 0.f, 0.f, 0.f, 0.f, 0.f};
#pragma unroll 4
    for (int h0 = 0; h0 < H; h0 += 4) {
        v2f a = *(const v2f*)(arow + h0);
        v2f b = *(const v2f*)(brow + h0);
        // (neg_a, A, neg_b, B, c_mod, C, reuse_a, reuse_b)
        c = __builtin_amdgcn_wmma_f32_16x16x4_f32(false, a, false, b,
                                                  (short)0, c, false, false);
    }

    // D layout: VGPR r -> lanes 0-15: M=r; lanes 16-31: M=8+r. All N equal.
    if (lane == 0) {
#pragma unroll
        for (int r = 0; r < 8; ++r) energies[s0 + r] = c[r];
    } else if (lane == 16) {
#pragma unroll
        for (int r = 0; r < 8; ++r) energies[s0 + 8 + r] = c[r];
    }
}

// ---- Stage 3: global max + sum(exp) (single block, deterministic) ---------
__global__ __launch_bounds__(1024) void k_stats(const float* __restrict__ energies,
                                                float* __restrict__ stats) {
    __shared__ float red[1024];
    const int t = threadIdx.x;

    float m = -3.4e38f;
    for (int i = t; i < S; i += 1024) m = fmaxf(m, energies[i]);
    red[t] = m;
    __syncthreads();
#pragma unroll
    for (int sft = 512; sft > 0; sft >>= 1) {
        if (t < sft) red[t] = fmaxf(red[t], red[t + sft]);
        __syncthreads();
    }
    const float gmax = red[0];
    __syncthreads();

    float acc = 0.0f;
    for (int i = t; i < S; i += 1024) acc += expf(energies[i] - gmax);
    red[t] = acc;
    __syncthreads();
#pragma unroll
    for (int sft = 512; sft > 0; sft >>= 1) {
        if (t < sft) red[t] += red[t + sft];
        __syncthreads();
    }
    if (t == 0) {
        stats[0] = gmax;
        stats[1] = 1.0f / red[0];
    }
}

// ---- Stage 4: normalize ---------------------------------------------------
__global__ __launch_bounds__(256) void k_norm(const float* __restrict__ energies,
                                              const float* __restrict__ stats,
                                              float* __restrict__ out) {
    const int i = blockIdx.x * 256 + threadIdx.x;
    out[i] = expf(energies[i] - stats[0]) * stats[1];
}

// ---------------------------------------------------------------------------
extern "C" void kernel_launch(void* const* d_in, const int* in_sizes, int n_in,
                              void* d_out, int out_size, void* d_ws, size_t ws_size,
                              hipStream_t stream) {
    // setup_inputs() order: hidden [H], encoder_outputs [S,H], W [H,H], b [H]
    const float* hidden = (const float*)d_in[0];
    const float* enc    = (const float*)d_in[1];
    const float* W      = (const float*)d_in[2];
    // b is unused: its contribution is constant over s and cancels in softmax.

    float* ws       = (float*)d_ws;
    float* part     = ws;                 // 16*H   floats
    float* v        = part + 16 * H;      // H      floats
    float* energies = v + H;              // S      floats
    float* stats    = energies + S;       // 2      floats
    float* out      = (float*)d_out;      // S floats (f32 output)

    k_vpart  <<<dim3(H / 256, 16), 256, 0, stream>>>(hidden, W, part);
    k_vreduce<<<H / 256,           256, 0, stream>>>(part, v);
    k_energy <<<S / 128,           256, 0, stream>>>(enc, v, energies);
    k_stats  <<<1,                1024, 0, stream>>>(energies, stats);
    k_norm   <<<S / 256,           256, 0, stream>>>(energies, stats, out);
}